// LSHSampledLayer_69965017252067
// MI455X (gfx1250) — compile-verified
//
#include <hip/hip_runtime.h>
#include <cstdint>
#include <cstddef>

// ---------------------------------------------------------------------------
// LSH sampled layer for MI455X (gfx1250, wave32).
//   logits[N,S] = x[N,D] @ Wg[S,D]^T + bg[S],  Wg/bg gathered by sample_ids.
// N=4096, D=128, S=32768.
//   - fp32 WMMA (V_WMMA_F32_16X16X4_F32): 34.4 GFLOP, ~16.8M WMMAs
//   - 512 MB logit stores -> ~22us HBM floor @ 23.3 TB/s -> NT store hint
//   - x (2MB) + gathered W (16MB) stay resident in 192MB L2 across blocks
//   - tile staging via GLOBAL_LOAD_ASYNC_TO_LDS_B128 (ASYNCcnt path)
// ---------------------------------------------------------------------------

typedef __attribute__((ext_vector_type(2))) float v2f;
typedef __attribute__((ext_vector_type(4))) float v4f;
typedef __attribute__((ext_vector_type(8))) float v8f;
typedef __attribute__((ext_vector_type(4))) int   v4i;

#define AS1 __attribute__((address_space(1)))
#define AS3 __attribute__((address_space(3)))

#if defined(__has_builtin)
#if __has_builtin(__builtin_amdgcn_global_load_async_to_lds_b128) && \
    __has_builtin(__builtin_amdgcn_s_wait_asynccnt)
#define USE_ASYNC_LDS 1
#endif
#endif

#define N_ROWS 4096
#define DIM    128
#define S_COLS 32768

#define BM  128          // block tile M
#define BS  128          // block tile S
#define LDA 132          // LDS row stride (floats); 132%64=4 -> conflict-free b64 frag loads

// ---- pass 1: gather sampled class rows + biases into dense workspace ------
__global__ void gather_rows(const float* __restrict__ weight,
                            const float* __restrict__ bias,
                            const long long* __restrict__ ids,
                            float* __restrict__ Wg,
                            float* __restrict__ bg) {
    // 256 threads: 8 rows/block, 32 threads/row, float4 per thread
    const int t    = threadIdx.x;
    const int rloc = t >> 5;            // 0..7
    const int c4   = t & 31;            // 0..31 float4 per 128-float row
    const int s    = blockIdx.x * 8 + rloc;
    const long long cls = ids[s];
    const v4f* src = (const v4f*)(weight + (size_t)cls * DIM);
    v4f*       dst = (v4f*)(Wg + (size_t)s * DIM);
    dst[c4] = src[c4];                  // regular store: keep Wg hot in L2
    if (c4 == 0) bg[s] = bias[cls];     // bias is [NUM_CLASS,1]
}

// ---- pass 2: sample_ids passthrough into output tail ----------------------
__global__ void copy_ids(const long long* __restrict__ ids,
                         long long* __restrict__ out) {
    const int i = blockIdx.x * 256 + threadIdx.x;
    out[i] = ids[i];
}

// ---- pass 3: fp32 WMMA GEMM -----------------------------------------------
// 256 threads = 8 waves.  Waves arranged 2(M) x 4(S); each wave owns a
// 64x32 tile = 4x2 accumulators of 16x16 f32 (8 VGPRs each).
__global__ void __launch_bounds__(256, 2)
lsh_gemm(const float* __restrict__ X,     // [N, D] row-major
         const float* __restrict__ Wg,    // [S, D] row-major (gathered)
         const float* __restrict__ bg,    // [S]
         float* __restrict__ out) {       // [N, S]
    extern __shared__ float lds[];
    float* ldsA = lds;                  // BM x K (stride LDA)
    float* ldsB = lds + BM * LDA;       // BS x K (stride LDA)

    const int t     = threadIdx.x;
    const int lane  = t & 31;
    const int wave  = t >> 5;
    const int waveM = wave & 1;         // 0..1
    const int waveS = wave >> 1;        // 0..3

    const int Mblk = blockIdx.y * BM;
    const int Sblk = blockIdx.x * BS;

    // Cooperative global->LDS stage of the full K=128 for both tiles.
    // 128 rows x 32 float4 = 4096 float4 per tile; 16 per thread.
#ifdef USE_ASYNC_LDS
    // CDNA5 async copy: no VGPR round-trip, tracked by ASYNCcnt.
    #pragma unroll
    for (int j = 0; j < 16; ++j) {
        const int i   = t + j * 256;
        const int row = i >> 5;
        const int c4  = (i & 31) * 4;
        __builtin_amdgcn_global_load_async_to_lds_b128(
            (AS1 v4i*)(X + (size_t)(Mblk + row) * DIM + c4),
            (AS3 v4i*)(ldsA + row * LDA + c4), 0, 0);
        __builtin_amdgcn_global_load_async_to_lds_b128(
            (AS1 v4i*)(Wg + (size_t)(Sblk + row) * DIM + c4),
            (AS3 v4i*)(ldsB + row * LDA + c4), 0, 0);
    }
    __builtin_amdgcn_s_wait_asynccnt(0);
#else
    #pragma unroll
    for (int j = 0; j < 16; ++j) {
        const int i   = t + j * 256;
        const int row = i >> 5;
        const int c4  = (i & 31) * 4;
        v4f a = *(const v4f*)(X  + (size_t)(Mblk + row) * DIM + c4);
        *(v4f*)(ldsA + row * LDA + c4) = a;
        v4f b = *(const v4f*)(Wg + (size_t)(Sblk + row) * DIM + c4);
        *(v4f*)(ldsB + row * LDA + c4) = b;
    }
#endif
    __syncthreads();

    // Bias folded into accumulator init: in the 16x16 f32 C/D layout the
    // column is N = lane&15 for every one of the 8 VGPRs, so a splat of
    // bg[col] is exactly "C = bias" before the K loop.
    const int colb = Sblk + waveS * 32 + (lane & 15);
    const float bv0 = bg[colb];
    const float bv1 = bg[colb + 16];
    v8f acc[4][2];
    #pragma unroll
    for (int mi = 0; mi < 4; ++mi) {
        #pragma unroll
        for (int r = 0; r < 8; ++r) { acc[mi][0][r] = bv0; acc[mi][1][r] = bv1; }
    }

    // ISA 16x4 f32 A-layout: lane&15 = matrix row, lane>>4 selects K-pair
    // {0,1} vs {2,3}; the v2f components land in consecutive VGPRs.
    const int   frow  = lane & 15;
    const int   khalf = (lane >> 4) * 2;
    const float* pa = ldsA + (waveM * 64 + frow) * LDA + khalf;
    const float* pb = ldsB + (waveS * 32 + frow) * LDA + khalf;

    #pragma unroll 8
    for (int ks = 0; ks < 32; ++ks) {           // K = 32 steps of 4
        v2f a[4], b[2];
        #pragma unroll
        for (int mi = 0; mi < 4; ++mi)
            a[mi] = *(const v2f*)(pa + mi * 16 * LDA + ks * 4);
        #pragma unroll
        for (int si = 0; si < 2; ++si)
            b[si] = *(const v2f*)(pb + si * 16 * LDA + ks * 4);
        #pragma unroll
        for (int mi = 0; mi < 4; ++mi)
            #pragma unroll
            for (int si = 0; si < 2; ++si)
                acc[mi][si] = __builtin_amdgcn_wmma_f32_16x16x4_f32(
                    /*neg_a=*/false, a[mi], /*neg_b=*/false, b[si],
                    /*c_mod=*/(short)0, acc[mi][si],
                    /*reuse_a=*/false, /*reuse_b=*/false);
    }

    // Epilogue: pure store (bias already in acc).  C/D layout: VGPR r holds
    // M = r + 8*(lane>>4), N = lane&15.  Each store = two aligned 64B
    // segments.  NT hint: logits are streamed once, keep L2 for x/Wg.
    const int rhalf = (lane >> 4) * 8;
    #pragma unroll
    for (int si = 0; si < 2; ++si) {
        const int col = colb + si * 16;
        #pragma unroll
        for (int mi = 0; mi < 4; ++mi) {
            const int rowb = Mblk + waveM * 64 + mi * 16 + rhalf;
            v8f c = acc[mi][si];
            #pragma unroll
            for (int r = 0; r < 8; ++r)
                __builtin_nontemporal_store(
                    c[r], &out[(size_t)(rowb + r) * S_COLS + col]);
        }
    }
}

extern "C" void kernel_launch(void* const* d_in, const int* in_sizes, int n_in,
                              void* d_out, int out_size, void* d_ws, size_t ws_size,
                              hipStream_t stream) {
    const float*     x      = (const float*)d_in[0];      // [N, D]
    const float*     weight = (const float*)d_in[1];      // [NUM_CLASS, D]
    const float*     bias   = (const float*)d_in[2];      // [NUM_CLASS, 1]
    const long long* ids    = (const long long*)d_in[3];  // [S] int64

    float* out = (float*)d_out;
    float* Wg  = (float*)d_ws;                       // [S, D]  16 MB
    float* bg  = Wg + (size_t)S_COLS * DIM;          // [S]     128 KB

    // 1) gather sampled rows + biases
    gather_rows<<<S_COLS / 8, 256, 0, stream>>>(weight, bias, ids, Wg, bg);

    // 2) ids passthrough (int64) appended after the N*S fp32 logits
    copy_ids<<<S_COLS / 256, 256, 0, stream>>>(
        ids, (long long*)(out + (size_t)N_ROWS * S_COLS));

    // 3) WMMA GEMM
    dim3 grid(S_COLS / BS, N_ROWS / BM);
    const size_t smem = (size_t)2 * BM * LDA * sizeof(float);  // ~135 KB of 320 KB
    lsh_gemm<<<grid, 256, smem, stream>>>(x, Wg, bg, out);
}